// SubNetwork_63608465654233
// MI455X (gfx1250) — compile-verified
//
#include <hip/hip_runtime.h>
#include <hip/hip_bf16.h>

#define N_NODES 100000
#define N_EDGES 1600000
#define IN_FEATS 128
#define HIDDEN 64
#define LAYERS 3
#define LN_EPS 1e-5f

typedef __attribute__((ext_vector_type(2))) float v2f;
typedef __attribute__((ext_vector_type(8))) float v8f;
typedef __attribute__((ext_vector_type(4))) unsigned int v4u;
typedef __attribute__((ext_vector_type(4))) int v4i;
typedef __attribute__((ext_vector_type(8))) int v8i;

// ---------------------------------------------------------------------------
// Degree flags: flag[n] = 1 iff node n has >=1 incoming edge.
// ---------------------------------------------------------------------------
__global__ void zero_flags_kernel(int* __restrict__ flags, int n) {
    int t = blockIdx.x * blockDim.x + threadIdx.x;
    if (t < n) flags[t] = 0;
}

__global__ void mark_flags_kernel(const int* __restrict__ dst,
                                  int* __restrict__ flags, int e) {
    int t = blockIdx.x * blockDim.x + threadIdx.x;
    if (t < e) flags[dst[t]] = 1;   // racy stores of identical value: benign
}

// ---------------------------------------------------------------------------
// Fused  h = relu(LN(vin @ W + b))  written to vout[:, 0:64];
// also zero-fills vout[:, 64:128] as the max-identity (messages are >= 0).
//
// One wave32 computes a 16x64 tile via V_WMMA_F32_16X16X4_F32 (full fp32).
// A 16x4 layout:  lane l (l15=l&15, half=l>>4): row=l15, K = 4*kk + 2*half + {0,1}
// B 4x16 layout:  col=l15, same K split per half
// C 16x16 layout: VGPR r -> row = r + 8*half, col = l15
//
// W (32 KB) is staged into LDS by the Tensor Data Mover: one
// tensor_load_to_lds per block (issued by wave 0, EXEC-independent),
// completion via s_wait_tensorcnt, visibility via the workgroup barrier.
// ---------------------------------------------------------------------------
#define WAVES_PER_BLOCK 8

__global__ void mlp_ln_relu_kernel(const float* __restrict__ vin,
                                   const float* __restrict__ W,      // [128][64]
                                   const float* __restrict__ bias,   // [64]
                                   const float* __restrict__ gamma,  // [64]
                                   const float* __restrict__ beta,   // [64]
                                   float* __restrict__ vout,         // [N][128]
                                   int ntiles) {
    __shared__ float sW[IN_FEATS * HIDDEN];   // W row-major: sW[k*64 + n]

#if __has_builtin(__builtin_amdgcn_tensor_load_to_lds)
    if (threadIdx.x < 32) {
        // Tensor DMA descriptor (D#), ISA ch.8: 1-D tile of 8192 x 4B.
        const unsigned lds_base = (unsigned)(size_t)(void*)sW;  // low 32 bits = LDS offset
        const unsigned long long ga = (unsigned long long)(size_t)W;
        v4u g0;
        g0.x = 1u;                                   // count=1, is_restore=0
        g0.y = lds_base;                             // lds_addr
        g0.z = (unsigned)(ga & 0xFFFFFFFFull);       // global_addr[31:0]
        g0.w = (unsigned)((ga >> 32) & 0x1FFFFFFull) // global_addr[56:32]
               | (2u << 30);                         // type=2 ("image")
        v8i g1;
        g1[0] = (2 << 16);          // workgroup_mask=0, data_size=2 (4 bytes)
        g1[1] = (8192 << 16);       // tensor_dim0[15:0] = 8192 (bits 63:48)
        g1[2] = (1 << 16);          // tensor_dim0[31:16]=0; tensor_dim1[15:0]=1
        g1[3] = (8192 << 16);       // tensor_dim1[31:16]=0; tile_dim0=8192
        g1[4] = 0;                  // tile_dim1=0 (unused), tile_dim2=0
        g1[5] = 8192;               // tensor_dim0_stride[31:0]
        g1[6] = 0;                  // stride hi, tensor_dim1_stride lo
        g1[7] = 0;
        v4i gz4 = {0, 0, 0, 0};         // groups 2/3 unused (<=2D tensor)
        v8i gz8 = {0, 0, 0, 0, 0, 0, 0, 0};
        __builtin_amdgcn_tensor_load_to_lds(g0, g1, gz4, gz4, gz8, 0);
        __builtin_amdgcn_s_wait_tensorcnt(0);
    }
#else
    for (int i = threadIdx.x; i < IN_FEATS * HIDDEN; i += blockDim.x) {
        sW[i] = W[i];
    }
#endif
    __syncthreads();

    const int wave = threadIdx.x >> 5;
    const int lane = threadIdx.x & 31;
    const int tile = blockIdx.x * WAVES_PER_BLOCK + wave;
    if (tile >= ntiles) return;

    const int half = lane >> 4;   // 0: lanes 0-15, 1: lanes 16-31
    const int l15  = lane & 15;

    const int rowA = tile * 16 + l15;
    const float* __restrict__ arow = vin + (size_t)rowA * IN_FEATS + half * 2;

    v8f acc[4];
#pragma unroll
    for (int nt = 0; nt < 4; ++nt) acc[nt] = (v8f)(0.0f);

#pragma unroll 4
    for (int kk = 0; kk < IN_FEATS / 4; ++kk) {
        const int k = kk * 4;
        v2f a;
        a.x = arow[k];
        a.y = arow[k + 1];
#pragma unroll
        for (int nt = 0; nt < 4; ++nt) {
            const int col = nt * 16 + l15;
            const float* bp = &sW[(k + half * 2) * HIDDEN + col];
            v2f b;
            b.x = bp[0];
            b.y = bp[HIDDEN];
            acc[nt] = __builtin_amdgcn_wmma_f32_16x16x4_f32(
                false, a, false, b, (short)0, acc[nt], false, false);
        }
    }

    // Per-column constants for this lane.
    float bi[4], gm[4], bt[4];
#pragma unroll
    for (int nt = 0; nt < 4; ++nt) {
        const int col = nt * 16 + l15;
        bi[nt] = bias[col];
        gm[nt] = gamma[col];
        bt[nt] = beta[col];
    }

    // Bias + LayerNorm (row-wise over 64 cols) + ReLU + store.
    // Row r's 64 values live in the 16 lanes of this lane's half across the
    // 4 accumulator tiles -> xor-shuffle masks 1,2,4,8 stay within the half.
#pragma unroll
    for (int r = 0; r < 8; ++r) {
        float hv[4];
#pragma unroll
        for (int nt = 0; nt < 4; ++nt) hv[nt] = acc[nt][r] + bi[nt];

        float s = hv[0] + hv[1] + hv[2] + hv[3];
        float q = hv[0] * hv[0] + hv[1] * hv[1] + hv[2] * hv[2] + hv[3] * hv[3];
#pragma unroll
        for (int m = 1; m < 16; m <<= 1) {
            s += __shfl_xor(s, m, 32);
            q += __shfl_xor(q, m, 32);
        }
        const float mu  = s * (1.0f / HIDDEN);
        const float var = q * (1.0f / HIDDEN) - mu * mu;
        const float inv = rsqrtf(var + LN_EPS);

        const int row = tile * 16 + r + half * 8;
        float* __restrict__ orow = vout + (size_t)row * (2 * HIDDEN);
#pragma unroll
        for (int nt = 0; nt < 4; ++nt) {
            const int col = nt * 16 + l15;
            float o = fmaf((hv[nt] - mu) * inv, gm[nt], bt[nt]);
            o = fmaxf(o, 0.0f);
            orow[col] = o;                 // h half of the concat
            orow[HIDDEN + col] = 0.0f;     // init agg half (max identity, h>=0)
        }
    }
}

// ---------------------------------------------------------------------------
// Edge scatter-max: agg[dst] = max(agg[dst], h[src]) per feature.
// h is post-ReLU (>= 0) so unsigned-int compare == float compare; u32 atomic
// max is deterministic and order-independent; agg (25.6 MB) lives in L2.
// ---------------------------------------------------------------------------
__global__ void scatter_max_kernel(const int* __restrict__ src,
                                   const int* __restrict__ dst,
                                   float* v,     // [N][128]: reads [:,0:64], atomics [:,64:128]
                                   int total) {
    int t = blockIdx.x * blockDim.x + threadIdx.x;
    if (t >= total) return;
    const int e = t >> 6;        // edge
    const int f = t & 63;        // feature
    const int s = src[e];
    const int d = dst[e];
    const float m = v[(size_t)s * 128 + f];
    atomicMax((unsigned int*)(v + (size_t)d * 128 + HIDDEN + f),
              __float_as_uint(m));
}

// ---------------------------------------------------------------------------
// Nodes with no incoming edges keep their own feature (DGL recv semantics).
// ---------------------------------------------------------------------------
__global__ void fixup_kernel(const int* __restrict__ flags,
                             float* __restrict__ v, int total) {
    int t = blockIdx.x * blockDim.x + threadIdx.x;
    if (t >= total) return;
    const int n = t >> 6;
    const int f = t & 63;
    if (flags[n] == 0) {
        v[(size_t)n * 128 + HIDDEN + f] = v[(size_t)n * 128 + f];
    }
}

// ---------------------------------------------------------------------------
extern "C" void kernel_launch(void* const* d_in, const int* in_sizes, int n_in,
                              void* d_out, int out_size, void* d_ws, size_t ws_size,
                              hipStream_t stream) {
    const float* inputs = (const float*)d_in[0];   // [N,128]
    const int*   src    = (const int*)d_in[1];     // [E]
    const int*   dst    = (const int*)d_in[2];     // [E]
    const float* Ws     = (const float*)d_in[3];   // [3,128,64]
    const float* bs     = (const float*)d_in[4];   // [3,64]
    const float* gammas = (const float*)d_in[5];   // [3,64]
    const float* betas  = (const float*)d_in[6];   // [3,64]
    float* out = (float*)d_out;                    // [N,128]

    float* vA    = (float*)d_ws;                   // [N,128]
    float* vB    = vA + (size_t)N_NODES * 128;     // [N,128]
    int*   flags = (int*)(vB + (size_t)N_NODES * 128);  // [N]

    // Degree flags (computed once, reused by every layer).
    zero_flags_kernel<<<(N_NODES + 255) / 256, 256, 0, stream>>>(flags, N_NODES);
    mark_flags_kernel<<<(N_EDGES + 255) / 256, 256, 0, stream>>>(dst, flags, N_EDGES);

    const int ntiles = N_NODES / 16;                       // 6250, exact
    const int mlp_blocks = (ntiles + WAVES_PER_BLOCK - 1) / WAVES_PER_BLOCK;
    const int sc_total = N_EDGES * 64;
    const int fx_total = N_NODES * 64;

    const float* vin = inputs;
    float* louts[LAYERS] = {vA, vB, out};
    for (int L = 0; L < LAYERS; ++L) {
        float* vo = louts[L];
        mlp_ln_relu_kernel<<<mlp_blocks, 32 * WAVES_PER_BLOCK, 0, stream>>>(
            vin, Ws + (size_t)L * IN_FEATS * HIDDEN, bs + L * HIDDEN,
            gammas + L * HIDDEN, betas + L * HIDDEN, vo, ntiles);
        scatter_max_kernel<<<(sc_total + 255) / 256, 256, 0, stream>>>(
            src, dst, vo, sc_total);
        fixup_kernel<<<(fx_total + 255) / 256, 256, 0, stream>>>(
            flags, vo, fx_total);
        vin = vo;
    }
}